// DiscoveryMemory_88596585382829
// MI455X (gfx1250) — compile-verified
//
#include <hip/hip_runtime.h>
#include <hip/hip_bf16.h>

// Problem constants (match reference)
#define B_      16
#define FEATS_  256
#define CODE_   128
#define HW_     16384
#define MEMSZ_  100
#define MPAD_   112       // memory rows padded to multiple of 16 for WMMA
#define DECAY_  0.9f

#define KSLAB_  64        // K columns of w_proj staged per LDS phase (kernel 1)
#define ASTRIDE_ 66       // KSLAB_+2 pad: 66*m mod 64 = 2m -> bank-conflict-free
#define MSTRIDE_ 130      // CODE_+2 pad for memory bank in LDS (kernel 4)

typedef __attribute__((ext_vector_type(2))) float v2f;
typedef __attribute__((ext_vector_type(8))) float v8f;

#if defined(__HIP_DEVICE_COMPILE__) && \
    !__has_builtin(__builtin_amdgcn_wmma_f32_16x16x4_f32)
#error "device pass: missing __builtin_amdgcn_wmma_f32_16x16x4_f32"
#endif

__device__ __forceinline__ v8f wmma4(v2f a, v2f b, v8f c) {
  // (neg_a, A, neg_b, B, c_mod, C, reuse_a, reuse_b)
  return __builtin_amdgcn_wmma_f32_16x16x4_f32(false, a, false, b, (short)0, c,
                                               false, false);
}

__device__ __forceinline__ v8f vzero8() {
  v8f z = {0.f, 0.f, 0.f, 0.f, 0.f, 0.f, 0.f, 0.f};
  return z;
}

// ---------------------------------------------------------------------------
// Kernel 1: proj[b,c,n] = sum_f w_proj[c,f]*feats[b,f,n] + b_proj[c]
// 8 waves/block; each wave owns one (b, 16-column) strip and all 8 M-tiles,
// so streaming feats is read exactly once from HBM. w_proj is staged into
// LDS in 64-wide K slabs shared by the whole block (8x less cache traffic),
// padded so A-fragment ds_load_b64 reads are bank-conflict-free.
// ---------------------------------------------------------------------------
__global__ void proj_gemm_kernel(const float* __restrict__ feats,
                                 const float* __restrict__ w_proj,
                                 const float* __restrict__ b_proj,
                                 float* __restrict__ out) {
  extern __shared__ float aslab[];  // CODE_ * ASTRIDE_ floats (33.8 KB)
  const int tid   = threadIdx.x;    // 256 threads = 8 waves
  const int wave  = (blockIdx.x * blockDim.x + tid) >> 5;
  const int lane  = tid & 31;
  const int b     = wave >> 10;     // HW_/16 = 1024 n-tiles per batch
  const int nT    = wave & 1023;
  const int col   = (nT << 4) + (lane & 15);
  const int khalf = (lane >> 4) << 1;  // lanes 0-15 -> K+0/1, 16-31 -> K+2/3

  const float* fb = feats + (long)b * FEATS_ * HW_;

  v8f acc[8];
#pragma unroll
  for (int i = 0; i < 8; ++i) acc[i] = vzero8();

  for (int kb = 0; kb < FEATS_; kb += KSLAB_) {
    __syncthreads();  // previous slab fully consumed
    for (int i = tid; i < CODE_ * KSLAB_; i += 256) {
      const int r = i >> 6, kk = i & (KSLAB_ - 1);
      aslab[r * ASTRIDE_ + kk] = w_proj[r * FEATS_ + kb + kk];
    }
    __syncthreads();

    for (int k = 0; k < KSLAB_; k += 4) {
      const int kg = kb + k;
      if (kg + 16 < FEATS_)  // stream hint for upcoming feats rows
        __builtin_prefetch(&fb[(long)(kg + 16 + khalf) * HW_ + col], 0, 0);
      v2f bf;
      bf.x = fb[(long)(kg + khalf) * HW_ + col];
      bf.y = fb[(long)(kg + khalf + 1) * HW_ + col];
#pragma unroll
      for (int ct = 0; ct < 8; ++ct) {
        const int m = (ct << 4) + (lane & 15);
        v2f af;
        af.x = aslab[m * ASTRIDE_ + k + khalf];
        af.y = aslab[m * ASTRIDE_ + k + khalf + 1];
        acc[ct] = wmma4(af, bf, acc[ct]);
      }
    }
  }

  float* ob = out + (long)b * (2 * CODE_) * HW_;  // proj = channels [0,128)
#pragma unroll
  for (int ct = 0; ct < 8; ++ct) {
#pragma unroll
    for (int j = 0; j < 8; ++j) {
      const int m = (ct << 4) + j + ((lane >> 4) << 3);
      ob[(long)m * HW_ + col] = acc[ct][j] + b_proj[m];
    }
  }
}

// ---------------------------------------------------------------------------
// Kernel 2: pooled[b,c] = mean_n proj[b,c,n]*preds[b,n]
// ---------------------------------------------------------------------------
__global__ void pooled_kernel(const float* __restrict__ out_proj,
                              const float* __restrict__ preds,
                              float* __restrict__ pooled) {
  const int bc = blockIdx.x;  // b*CODE_ + c
  const int b = bc >> 7, c = bc & 127;
  const float* p  = out_proj + (long)b * 2 * CODE_ * HW_ + (long)c * HW_;
  const float* pr = preds + (long)b * HW_;
  float s = 0.f;
  for (int n = threadIdx.x; n < HW_; n += blockDim.x) s += p[n] * pr[n];
  __shared__ float red[256];
  red[threadIdx.x] = s;
  __syncthreads();
  for (int off = 128; off > 0; off >>= 1) {
    if ((int)threadIdx.x < off) red[threadIdx.x] += red[threadIdx.x + off];
    __syncthreads();
  }
  if (threadIdx.x == 0) pooled[bc] = red[0] * (1.0f / HW_);
}

// ---------------------------------------------------------------------------
// Kernel 3: sequential EMA-or-append scan (single workgroup, 128 threads).
// Works on a workspace copy of memory (inputs are read-only).
// ---------------------------------------------------------------------------
__global__ void mem_update_kernel(const float* __restrict__ mem_in,
                                  const int* __restrict__ ptr_in,
                                  const float* __restrict__ pooled,
                                  float* __restrict__ mem_ws,
                                  int* __restrict__ ptr_ws) {
  const int tid = threadIdx.x;  // 128 threads == CODE_
  for (int i = tid; i < MEMSZ_ * CODE_; i += 128) mem_ws[i] = mem_in[i];
  __shared__ float sims[128];
  __shared__ float red[128];
  __shared__ int s_idx, s_ema, s_p;
  if (tid == 0) s_p = ptr_in[0];
  __syncthreads();

  for (int v = 0; v < B_; ++v) {
    const float* vec = pooled + v * CODE_;
    const float x = vec[tid];
    red[tid] = x * x;
    __syncthreads();
    for (int off = 64; off > 0; off >>= 1) {
      if (tid < off) red[tid] += red[tid + off];
      __syncthreads();
    }
    const float vnorm = sqrtf(red[0]);
    const int p = s_p;
    __syncthreads();

    if (tid < MEMSZ_) {
      float dot = 0.f, rn2 = 0.f;
      const float* row = mem_ws + tid * CODE_;
      for (int c = 0; c < CODE_; ++c) {
        const float rv = row[c];
        dot += rv * vec[c];
        rn2 += rv * rv;
      }
      const float rn  = sqrtf(rn2);
      const float sim = (dot / (rn == 0.f ? 1.f : rn)) / vnorm;
      sims[tid] = (tid < p) ? sim : -2.0f;
    }
    __syncthreads();

    if (tid == 0) {
      int best = 0;
      float bv = sims[0];
      for (int m = 1; m < MEMSZ_; ++m)
        if (sims[m] > bv) { bv = sims[m]; best = m; }
      s_idx = best;
      s_ema = (p > 0 && bv >= 0.5f) ? 1 : 0;
    }
    __syncthreads();

    if (s_ema) {
      float* row = mem_ws + s_idx * CODE_;
      row[tid] = row[tid] * DECAY_ + (1.0f - DECAY_) * x;
    } else {
      mem_ws[p * CODE_ + tid] = x;
    }
    __syncthreads();
    if (tid == 0 && !s_ema) s_p = p + 1;
    __syncthreads();
  }
  if (tid == 0) ptr_ws[0] = s_p;
}

// ---------------------------------------------------------------------------
// Kernel 4: attention.  Block = (b, 128-column tile), 8 waves.
// Memory bank (zero-padded to 112 rows) is staged once into LDS; logits are
// computed by WMMA into a second LDS region; masked softmax per column;
// feats_aug = memory^T x attn by WMMA; store to channels [128,256).
// Dynamic LDS: 112*130 + 112*128 floats = 113 KB (2 blocks / WGP).
// ---------------------------------------------------------------------------
__global__ void attn_kernel(const float* __restrict__ out_proj,
                            const float* __restrict__ mem_ws,
                            const int* __restrict__ ptr_ws,
                            float* __restrict__ out) {
  extern __shared__ float lds[];
  float* memL = lds;                     // MPAD_ * MSTRIDE_ floats
  float* slog = lds + MPAD_ * MSTRIDE_;  // MPAD_ * 128 floats
  const int tid = threadIdx.x;
  const int wave = tid >> 5, lane = tid & 31;
  const int b = blockIdx.x >> 7;  // 128 column-tiles per batch
  const int nBase = (blockIdx.x & 127) * 128;
  const int khalf = (lane >> 4) << 1;
  const int ptr = ptr_ws[0];

  // Stage memory bank into LDS (rows >= MEMSZ_ zeroed: no guards in GEMMs)
  for (int i = tid; i < MPAD_ * CODE_; i += 256) {
    const int r = i >> 7, c = i & 127;
    memL[r * MSTRIDE_ + c] = (r < MEMSZ_) ? mem_ws[r * CODE_ + c] : 0.0f;
  }
  __syncthreads();

  const float* proj = out_proj + (long)b * 2 * CODE_ * HW_;
  const int nl = (wave << 4) + (lane & 15);  // local column within tile
  const int nw = nBase + nl;                 // global pixel column

  // --- logits GEMM: each wave owns a 16-col strip, all 7 M-tiles ---
  v8f acc[7];
#pragma unroll
  for (int i = 0; i < 7; ++i) acc[i] = vzero8();
  for (int k = 0; k < CODE_; k += 4) {
    v2f bf;
    bf.x = proj[(long)(k + khalf) * HW_ + nw];
    bf.y = proj[(long)(k + khalf + 1) * HW_ + nw];
#pragma unroll
    for (int mt = 0; mt < 7; ++mt) {
      const int m = (mt << 4) + (lane & 15);
      v2f af;
      af.x = memL[m * MSTRIDE_ + k + khalf];
      af.y = memL[m * MSTRIDE_ + k + khalf + 1];
      acc[mt] = wmma4(af, bf, acc[mt]);
    }
  }
#pragma unroll
  for (int mt = 0; mt < 7; ++mt) {
#pragma unroll
    for (int j = 0; j < 8; ++j) {
      const int m = (mt << 4) + j + ((lane >> 4) << 3);
      slog[m * 128 + nl] = acc[mt][j];
    }
  }
  __syncthreads();

  // --- masked softmax over m, one thread per column ---
  if (tid < 128) {
    float mx = -3.4e38f;
    for (int m = 0; m < ptr; ++m) mx = fmaxf(mx, slog[m * 128 + tid]);
    float sum = 0.f;
    for (int m = 0; m < ptr; ++m) sum += __expf(slog[m * 128 + tid] - mx);
    const float inv = 1.0f / sum;
    for (int m = 0; m < MPAD_; ++m) {
      const float a =
          (m < ptr) ? __expf(slog[m * 128 + tid] - mx) * inv : 0.0f;
      slog[m * 128 + tid] = a;  // rows >= ptr zeroed: GEMM2 needs no mask
    }
  }
  __syncthreads();

  // --- feats_aug GEMM: A[c,m] = memory[m,c], B[m,n] = attn ---
  v8f acc2[8];
#pragma unroll
  for (int i = 0; i < 8; ++i) acc2[i] = vzero8();
  for (int k = 0; k < MPAD_; k += 4) {
    v2f bf;
    bf.x = slog[(k + khalf) * 128 + nl];
    bf.y = slog[(k + khalf + 1) * 128 + nl];
#pragma unroll
    for (int ct = 0; ct < 8; ++ct) {
      const int c = (ct << 4) + (lane & 15);
      v2f af;
      af.x = memL[(k + khalf) * MSTRIDE_ + c];
      af.y = memL[(k + khalf + 1) * MSTRIDE_ + c];
      acc2[ct] = wmma4(af, bf, acc2[ct]);
    }
  }
  float* ob = out + (long)b * 2 * CODE_ * HW_ + (long)CODE_ * HW_;
#pragma unroll
  for (int ct = 0; ct < 8; ++ct) {
#pragma unroll
    for (int j = 0; j < 8; ++j) {
      const int c = (ct << 4) + j + ((lane >> 4) << 3);
      ob[(long)c * HW_ + nw] = acc2[ct][j];
    }
  }
}

// ---------------------------------------------------------------------------
extern "C" void kernel_launch(void* const* d_in, const int* in_sizes, int n_in,
                              void* d_out, int out_size, void* d_ws,
                              size_t ws_size, hipStream_t stream) {
  const float* feats  = (const float*)d_in[0];
  const float* preds  = (const float*)d_in[1];
  const float* w_proj = (const float*)d_in[2];
  const float* b_proj = (const float*)d_in[3];
  const float* memory = (const float*)d_in[4];
  const int*   ptr    = (const int*)d_in[5];
  float* out = (float*)d_out;

  // workspace layout
  float* pooled = (float*)d_ws;                     // B*CODE floats
  float* mem_ws = pooled + B_ * CODE_;              // MEMSZ*CODE floats
  int*   ptr_ws = (int*)(mem_ws + MEMSZ_ * CODE_);  // 1 int

  const size_t lds1 = (size_t)CODE_ * ASTRIDE_ * sizeof(float);         // 33.8 KB
  const size_t lds4 =
      ((size_t)MPAD_ * MSTRIDE_ + (size_t)MPAD_ * 128) * sizeof(float); // 113 KB

  // K1: 16384 waves = B * (HW/16); 8 waves per block
  proj_gemm_kernel<<<2048, 256, lds1, stream>>>(feats, w_proj, b_proj, out);
  // K2: one block per (b,c)
  pooled_kernel<<<B_ * CODE_, 256, 0, stream>>>(out, preds, pooled);
  // K3: single workgroup sequential scan
  mem_update_kernel<<<1, 128, 0, stream>>>(memory, ptr, pooled, mem_ws, ptr_ws);
  // K4: one block per (b, 128-column tile)
  attn_kernel<<<B_ * 128, 256, lds4, stream>>>(out, mem_ws, ptr_ws, out);
}